// HyperbolicClassCentroids_5746666242182
// MI455X (gfx1250) — compile-verified
//
#include <hip/hip_runtime.h>
#include <hip/hip_bf16.h>

// ---------------- problem constants (from reference) ----------------
#define D1      129           // 1 time + 128 spatial components
#define ACC_STR 130           // per-class accumulator stride: 129 comps + 1 count
#define NR      8             // class ranges
#define CR      125           // classes per range (NR*CR = 1000)
#define BLOCK_B 512           // 16 waves
#define WAVES_B (BLOCK_B / 32)
#define NCHUNK  120           // point chunks -> grid = 120 x 8 = 960 blocks
#define TBATCH  8             // rows per TDM gather (32-bit indices => max 8)
#define EPSF    1e-7f

typedef unsigned int u32x4 __attribute__((ext_vector_type(4)));
typedef int          i32x8 __attribute__((ext_vector_type(8)));
typedef int          i32x4 __attribute__((ext_vector_type(4)));
typedef float        v2f   __attribute__((ext_vector_type(2)));
typedef float        v8f   __attribute__((ext_vector_type(8)));

// full-wave (wave32) butterfly sum: every lane ends with the total
__device__ __forceinline__ float wsum(float v) {
#pragma unroll
    for (int m = 16; m > 0; m >>= 1) v += __shfl_xor(v, m, 32);
    return v;
}

// TDM gather: DMA `nrows` scattered 129-float rows of z into contiguous LDS.
// D# built per CDNA5 ISA 8.3/8.4/8.7 (gather mode, 32-bit indices).
__device__ __forceinline__ void tdm_gather_rows(const float* zbase, unsigned ldsOff,
                                                int nrows, unsigned npts,
                                                int r0, int r1, int r2, int r3,
                                                int r4, int r5, int r6, int r7)
{
    const unsigned long long ga = (unsigned long long)(size_t)zbase;
    u32x4 g0;
    g0.x = 1u | (1u << 30) | (1u << 31);              // count=1, idx=32b, gather_mode=1
    g0.y = ldsOff;                                    // LDS byte offset (wave-relative)
    g0.z = (unsigned)(ga & 0xFFFFFFFFu);              // global_addr[31:0]
    g0.w = (unsigned)((ga >> 32) & 0x01FFFFFFu)       // global_addr[56:32]
         | (2u << 30);                                // type=2 ("image")
    i32x8 g1;
    g1[0] = (int)(2u << 16);                          // workgroup_mask=0, data_size=4B
    g1[1] = (int)((unsigned)(D1 & 0xFFFF) << 16);     // tensor_dim0[15:0]=129
    g1[2] = (int)((npts & 0xFFFFu) << 16);            // dim0[31:16]=0 | tensor_dim1[15:0]
    g1[3] = (int)((npts >> 16) & 0xFFFFu)             // tensor_dim1[31:16]
          | (int)((unsigned)D1 << 16);                // tile_dim0 = 129
    g1[4] = nrows & 0xFFFF;                           // tile_dim1 = #valid indices
    g1[5] = D1;                                       // tensor_dim0_stride[31:0] = 129
    g1[6] = 0;                                        // stride[47:32]=0 (dim1_stride ignored)
    g1[7] = 0;
    i32x4 g2 = { r0, r1, r2, r3 };                    // row indices 0..3 (32-bit mode)
    i32x4 g3 = { r4, r5, r6, r7 };                    // row indices 4..7
    const i32x8 g4 = { 0, 0, 0, 0, 0, 0, 0, 0 };      // unused VADDR4 slot (zero-filled)
    __builtin_amdgcn_tensor_load_to_lds(g0, g1, g2, g3, g4, 0);
}

// ---------------------------------------------------------------
// Pass 1: per-point log-map + class-partitioned LDS segment sum.
// grid.x = point chunk, grid.y = class range (CR classes -> 65 KB LDS)
// Each wave scans 32 candidate points at a time (coalesced y reads),
// ballots the in-range ones, TDM-gathers their z rows into LDS, then
// processes one point per wave iteration (lane L owns components
// L, L+32, L+64, L+96; lane 0 owns component 128 + Lorentz sign fix).
// ---------------------------------------------------------------
__global__ __launch_bounds__(BLOCK_B) void centroid_pass(
    const float* __restrict__ z, const float* __restrict__ centers,
    const int* __restrict__ y, float* __restrict__ gacc, int npts)
{
    extern __shared__ float smem[];
    float* lacc  = smem;                                     // CR*ACC_STR accumulator
    const int lane  = threadIdx.x & 31;
    const int wave  = threadIdx.x >> 5;
    const int range = blockIdx.y;
    const int cbase = range * CR;
    float* stage = smem + CR * ACC_STR + wave * (TBATCH * D1);   // per-wave TDM target
    const unsigned stageOff = (unsigned)(size_t)(const void*)stage; // LDS byte offset

    for (int i = threadIdx.x; i < CR * ACC_STR; i += BLOCK_B) lacc[i] = 0.0f;
    __syncthreads();

    const int per = (npts + gridDim.x - 1) / gridDim.x;
    const int p0  = blockIdx.x * per;
    const int p1  = min(npts, p0 + per);

    for (int base = p0 + wave * 32; base < p1; base += WAVES_B * 32) {
        __builtin_prefetch(y + base + WAVES_B * 64, 0, 0);        // global_prefetch_b8
        const int ptl = base + lane;                               // coalesced y scan
        const int yvl = (ptl < p1) ? y[ptl] : -1;
        const bool match = ((unsigned)(yvl - cbase) < (unsigned)CR);
        unsigned mask = (unsigned)__ballot(match);

        while (mask) {
            unsigned mm = mask;
            int b[TBATCH];
#pragma unroll
            for (int j = 0; j < TBATCH; ++j) {
                b[j] = mm ? (__ffs(mm) - 1) : 0;                   // repeat-safe filler
                mm &= mm - 1;
            }
            int n = __popc(mask);
            if (n > TBATCH) n = TBATCH;
            mask = mm;                                             // bits left for next batch

            // one wave-level DMA for up to 8 scattered z rows -> contiguous LDS
            tdm_gather_rows(z, stageOff, n, (unsigned)npts,
                            base + b[0], base + b[1], base + b[2], base + b[3],
                            base + b[4], base + b[5], base + b[6], base + b[7]);
            __builtin_amdgcn_s_wait_tensorcnt(0);
            __asm__ __volatile__("" ::: "memory");                 // don't hoist LDS reads

#pragma unroll
            for (int j = 0; j < TBATCH; ++j) {
                if (j >= n) break;
                const int yv = __shfl(yvl, b[j], 32);
                const float* zr = stage + j * D1;                  // LDS (TDM-gathered)
                const float* pr = centers + (size_t)yv * D1;       // hot in L2 (516 KB)

                float zv[4], pv[4];
#pragma unroll
                for (int k = 0; k < 4; ++k) {
                    zv[k] = zr[lane + 32 * k];
                    pv[k] = pr[lane + 32 * k];
                }
                float z8 = 0.0f, p8 = 0.0f;
                if (lane == 0) { z8 = zr[128]; p8 = pr[128]; }

                // inner_pq = <p, z>_L
                float part = 0.0f;
#pragma unroll
                for (int k = 0; k < 4; ++k) part += zv[k] * pv[k];
                if (lane == 0) part += z8 * p8 - 2.0f * zv[0] * pv[0];
                const float ip = wsum(part);

                const float alpha = fmaxf(-ip, 1.0f + EPSF);
                const float dist  = logf(alpha + sqrtf(fmaxf(alpha * alpha - 1.0f, 0.0f)));

                // direction = z + ip * p
                float dv[4];
#pragma unroll
                for (int k = 0; k < 4; ++k) dv[k] = fmaf(ip, pv[k], zv[k]);
                float d8 = fmaf(ip, p8, z8);

                // c2 = <p, direction>_L ; direction += c2 * p
                part = 0.0f;
#pragma unroll
                for (int k = 0; k < 4; ++k) part += pv[k] * dv[k];
                if (lane == 0) part += p8 * d8 - 2.0f * pv[0] * dv[0];
                const float c2 = wsum(part);
#pragma unroll
                for (int k = 0; k < 4; ++k) dv[k] = fmaf(c2, pv[k], dv[k]);
                d8 = fmaf(c2, p8, d8);

                // a3 = <dir,dir>_L ; a4 = <p,dir>_L
                float pa = 0.0f, pb = 0.0f;
#pragma unroll
                for (int k = 0; k < 4; ++k) { pa += dv[k] * dv[k]; pb += pv[k] * dv[k]; }
                if (lane == 0) {
                    pa += d8 * d8 - 2.0f * dv[0] * dv[0];
                    pb += p8 * d8 - 2.0f * pv[0] * dv[0];
                }
                const float a3 = wsum(pa);
                const float a4 = wsum(pb);

                const float dn = sqrtf(fmaxf(a3, EPSF));
                const float f  = dist / dn;
                const float cp = f * a4;       // v = f*dir + (f*a4)*p

                const unsigned lcl = (unsigned)(yv - cbase);
                float* slot = lacc + (size_t)lcl * ACC_STR;
#pragma unroll
                for (int k = 0; k < 4; ++k)
                    atomicAdd(slot + lane + 32 * k, fmaf(cp, pv[k], f * dv[k])); // ds_add_f32
                if (lane == 0) {
                    atomicAdd(slot + 128, fmaf(cp, p8, f * d8));
                    atomicAdd(slot + 129, 1.0f);                                 // count
                }
            }
        }
    }

    __syncthreads();
    // Dump this block's partial into the global class accumulator.
    float* gdst = gacc + (size_t)cbase * ACC_STR;
    for (int i = threadIdx.x; i < CR * ACC_STR; i += BLOCK_B) {
        const float vv = lacc[i];
        if (vv != 0.0f)
            __hip_atomic_fetch_add(gdst + i, vv, __ATOMIC_RELAXED,
                                   __HIP_MEMORY_SCOPE_AGENT);       // global_atomic_add_f32
    }
}

// extract Gram diagonal [j][j] from a 16x16 f32 WMMA accumulator:
// element lives in VGPR (j&7), lane (j<8 ? j : j+16)
__device__ __forceinline__ float diag16(const v8f& c, int j) {
    return __shfl(c[j & 7], (j < 8) ? j : (j + 16), 32);
}

// ---------------------------------------------------------------
// Pass 2: per-class exp-map update. One wave per 16 classes.
// The three Lorentz inner products per class are computed as the
// diagonals of three 16x16 Gram matrices via v_wmma_f32_16x16x4_f32
// chains over K=129 (A = comp0-negated operand); everything else
// follows by exact bilinear expansion, so no shuffle reductions.
// ---------------------------------------------------------------
__global__ __launch_bounds__(256) void centroid_finalize(
    const float* __restrict__ centers, const unsigned char* __restrict__ initialized,
    const float* __restrict__ gacc, float* __restrict__ out, int ncls)
{
    const int lane   = threadIdx.x & 31;
    const int wave   = threadIdx.x >> 5;
    const int ntiles = (ncls + 15) >> 4;
    const int tile   = blockIdx.x * (blockDim.x >> 5) + wave;
    if (tile >= ntiles) return;                       // wave-uniform: EXEC stays all-1s

    // lane -> class row m (both lane halves mirror rows 0..15)
    const int m   = lane & 15;
    int clm       = tile * 16 + m;
    if (clm >= ncls) clm = ncls - 1;                  // clamp loads; writes guarded later
    const float cntm = gacc[(size_t)clm * ACC_STR + 129];
    const float sclm = (initialized[clm] ? 0.1f : 1.0f) / fmaxf(cntm, 1.0f);

    const float* crow = centers + (size_t)clm * D1;
    const float* arow = gacc + (size_t)clm * ACC_STR;

    // Gram accumulators: c1 = P~ U^T, c2 = U~ U^T, c3 = P~ P^T
    v8f c1 = {}, c2 = {}, c3 = {};
    const int khalf = 2 * (lane >> 4);                // K split across lane halves
    for (int kc = 0; kc <= 128; kc += 4) {
        const int ks = kc + khalf;
        float pa = 0.0f, pb = 0.0f, ua = 0.0f, ub = 0.0f;
        if (ks <= 128)     { pa = crow[ks];     ua = arow[ks]     * sclm; }
        if (ks + 1 <= 128) { pb = crow[ks + 1]; ub = arow[ks + 1] * sclm; }
        const float sgn = (ks == 0) ? -1.0f : 1.0f;   // Lorentz sign on A operand
        const v2f Ap = { pa * sgn, pb };
        const v2f Au = { ua * sgn, ub };
        const v2f Bp = { pa, pb };
        const v2f Bu = { ua, ub };
        c1 = __builtin_amdgcn_wmma_f32_16x16x4_f32(false, Ap, false, Bu, (short)0, c1, false, false);
        c2 = __builtin_amdgcn_wmma_f32_16x16x4_f32(false, Au, false, Bu, (short)0, c2, false, false);
        c3 = __builtin_amdgcn_wmma_f32_16x16x4_f32(false, Ap, false, Bp, (short)0, c3, false, false);
    }

    // serial per-class write-out (no reductions needed)
#pragma unroll
    for (int j = 0; j < 16; ++j) {
        const int cl = tile * 16 + j;
        if (cl >= ncls) break;
        const float a1  = diag16(c1, j);              // <p,u>_L
        const float iuu = diag16(c2, j);              // <u,u>_L
        const float ipp = diag16(c3, j);              // <p,p>_L
        const float cnt = __shfl(cntm, j, 32);
        const float scl = __shfl(sclm, j, 32);

        const float a2 = iuu + a1 * a1 * (2.0f + ipp);          // <vt,vt>_L
        const float vn = sqrtf(fmaxf(a2, EPSF));
        const float ch = coshf(vn);
        const float sh = sinhf(vn) / vn;

        const float* pj = centers + (size_t)cl * D1;
        const float* aj = gacc + (size_t)cl * ACC_STR;
        const float p0  = pj[0];
        const float u0  = aj[0] * scl;
        const float o0  = ch * p0 + sh * (u0 + a1 * p0);

        // sum_{c>=1} out_c^2 = <out,out>_L + out0^2 (exact bilinear expansion)
        const float innLoo = ch * ch * ipp
                           + 2.0f * ch * sh * a1 * (1.0f + ipp)
                           + sh * sh * a2;
        const float tim = sqrtf(fmaxf(1.0f + innLoo + o0 * o0, EPSF));

        const bool upd = cnt > 0.0f;
        float* orow = out + (size_t)cl * D1;
#pragma unroll
        for (int k = 0; k < 4; ++k) {
            const int c   = lane + 32 * k;
            const float p = pj[c];
            const float u = aj[c] * scl;
            const float o = ch * p + sh * fmaf(a1, p, u);
            orow[c] = upd ? (c == 0 ? tim : o) : p;
        }
        if (lane == 0) {
            const float p = pj[128];
            const float u = aj[128] * scl;
            const float o = ch * p + sh * fmaf(a1, p, u);
            orow[128] = upd ? o : p;
        }
    }
}

// ---------------------------------------------------------------
extern "C" void kernel_launch(void* const* d_in, const int* in_sizes, int n_in,
                              void* d_out, int out_size, void* d_ws, size_t ws_size,
                              hipStream_t stream)
{
    const float*         z    = (const float*)d_in[0];
    const float*         cen  = (const float*)d_in[1];
    const int*           y    = (const int*)d_in[2];
    const unsigned char* init = (const unsigned char*)d_in[3];
    float*               outp = (float*)d_out;
    float*               gacc = (float*)d_ws;

    const int npts = in_sizes[0] / D1;     // 524288
    const int ncls = in_sizes[1] / D1;     // 1000

    const size_t accBytes = (size_t)ncls * ACC_STR * sizeof(float);   // 520 KB
    (void)hipMemsetAsync(d_ws, 0, accBytes, stream);

    const size_t shmB = (size_t)(CR * ACC_STR + WAVES_B * TBATCH * D1) * sizeof(float);
    dim3 gridB(NCHUNK, NR);
    centroid_pass<<<gridB, BLOCK_B, shmB, stream>>>(z, cen, y, gacc, npts);

    const int ntiles  = (ncls + 15) / 16;            // 63
    const int blocksC = (ntiles + 7) / 8;            // 8 waves (tiles) per block
    centroid_finalize<<<blocksC, 256, 0, stream>>>(cen, init, gacc, outp, ncls);
}